// DSNTLI_19207093748475
// MI455X (gfx1250) — compile-verified
//
#include <hip/hip_runtime.h>

// ---------------------------------------------------------------------------
// DSNT (expected-coordinate GEMV) + leaky-integrator scan for MI455X (gfx1250)
//
// Bandwidth-bound: 629 MB of x streamed once -> ~27 us floor @ 23.3 TB/s.
// Kernel 1 streams x with b128 non-temporal loads and reduces per-wave partial
// sums with V_WMMA_F32_16X16X4_F32 (matrix engine used as a 32-lane reducer).
// Kernel 2 combines K-slice partials and runs the tiny sequential scan.
// ---------------------------------------------------------------------------

typedef __attribute__((ext_vector_type(4))) float f4;
typedef __attribute__((ext_vector_type(2))) float v2f;
typedef __attribute__((ext_vector_type(8))) float v8f;

#define T_DIM   512
#define H_DIM   480
#define W_DIM   640
#define ROW_ELEMS   (H_DIM * W_DIM)      // 307200
#define ROW_F4      (ROW_ELEMS / 4)      // 76800 float4 per frame
#define NSLICE      10                   // K-split per frame
#define SLICE_F4    (ROW_F4 / NSLICE)    // 7680 float4 per slice
#define ITERS       (SLICE_F4 / 256)     // 30 float4 per thread

// Partial-sum kernel: block (t, s) reduces slice s of frame t.
// ws layout: ws[(t*NSLICE + s)*2 + ch], ch0 = y-dot (co_2), ch1 = x-dot (co_1)
__global__ __launch_bounds__(256) void dsnt_partial_kernel(
    const float* __restrict__ x, float* __restrict__ ws) {
  const int t = blockIdx.x;
  const int s = blockIdx.y;
  const int tid = threadIdx.x;
  const int lane = tid & 31;
  const int wave = tid >> 5;

  const f4* __restrict__ row = (const f4*)(x + (size_t)t * ROW_ELEMS);
  const int base = s * SLICE_F4;

  const float cX = 2.0f / (float)(W_DIM - 1);
  const float cY = 2.0f / (float)(H_DIM - 1);

  float sx = 0.0f;  // dot with probs_x
  float sy = 0.0f;  // dot with probs_y

#pragma unroll 6
  for (int it = 0; it < ITERS; ++it) {
    const int fidx = base + it * 256 + tid;           // float4 index in frame
    const f4 v = __builtin_nontemporal_load(row + fidx);
    // W=640 is a multiple of 4 -> all 4 elements share one image row.
    const int irow = fidx / (W_DIM / 4);              // image row (0..479)
    const int j0 = (fidx - irow * (W_DIM / 4)) * 4;   // first column (0..636)
    const float sv = (v[0] + v[1]) + (v[2] + v[3]);
    const float tw = v[1] + 2.0f * v[2] + 3.0f * v[3];
    // sum_e v_e * (cX*(j0+e) - 1) = cX*(j0*sv + tw) - sv
    sx = fmaf(cX, fmaf((float)j0, sv, tw), sx) - sv;
    // sum_e v_e * (cY*irow - 1)   = cY*irow*sv - sv
    sy = fmaf(cY * (float)irow, sv, sy) - sv;
  }

  // ---- wave32 reduction via V_WMMA_F32_16X16X4_F32 ----
  // A (16x4, 2 VGPRs): lanes 0-15 hold A[l,0]=sx_l, A[l,1]=sy_l;
  //                    lanes 16-31 hold A[l-16,2]=sx_l, A[l-16,3]=sy_l.
  // B (4x16, 2 VGPRs): VGPR0 = rows K0 (lanes 0-15) / K2 (lanes 16-31),
  //                    VGPR1 = rows K1/K3. Ones in column0 of K0/K2 gather sx,
  //                    ones in column1 of K1/K3 gather sy.
  v2f av; av[0] = sx; av[1] = sy;
  v2f bv;
  bv[0] = (lane == 0 || lane == 16) ? 1.0f : 0.0f;
  bv[1] = (lane == 1 || lane == 17) ? 1.0f : 0.0f;
  v8f cz = {};
  v8f d = __builtin_amdgcn_wmma_f32_16x16x4_f32(
      /*neg_a=*/false, av, /*neg_b=*/false, bv,
      /*c_mod=*/(short)0, cz, /*reuse_a=*/false, /*reuse_b=*/false);

  // Per-lane sum over the 8 C/D VGPRs: lane n<16 -> sum_M=0..7 D[M,n],
  // lane n>=16 -> sum_M=8..15 D[M,n-16].
  float p = ((d[0] + d[1]) + (d[2] + d[3])) + ((d[4] + d[5]) + (d[6] + d[7]));
  // Column 0 (= total sx) lives in lanes 0 and 16; column 1 (sy) in 1 and 17.
  const float sx_tot = __shfl(p, 0, 32) + __shfl(p, 16, 32);
  const float sy_tot = __shfl(p, 1, 32) + __shfl(p, 17, 32);

  __shared__ float red[8][2];
  if (lane == 0) {
    red[wave][0] = sy_tot;  // channel 0 = co_2 (y)
    red[wave][1] = sx_tot;  // channel 1 = co_1 (x)
  }
  __syncthreads();

  if (tid == 0) {
    float a0 = 0.0f, a1 = 0.0f;
#pragma unroll
    for (int w = 0; w < 8; ++w) { a0 += red[w][0]; a1 += red[w][1]; }
    float* dst = ws + ((size_t)t * NSLICE + s) * 2;
    dst[0] = a0;
    dst[1] = a1;
  }
}

// Combine K-slice partials, then run the sequential leaky-integrator scan.
// out layout: out[t*2+ch] for t=0..511, then out[1024..1025] = final state.
__global__ __launch_bounds__(256) void dsnt_scan_kernel(
    const float* __restrict__ ws, const float* __restrict__ li_tm,
    const float* __restrict__ state0, float* __restrict__ out) {
  __shared__ float cos_sh[T_DIM * 2];

  for (int idx = threadIdx.x; idx < T_DIM * 2; idx += blockDim.x) {
    const int t = idx >> 1;
    const int ch = idx & 1;
    float acc = 0.0f;
#pragma unroll
    for (int s = 0; s < NSLICE; ++s)
      acc += ws[((size_t)t * NSLICE + s) * 2 + ch];
    cos_sh[idx] = acc;
  }
  __syncthreads();

  if (threadIdx.x < 2) {
    const int ch = threadIdx.x;
    const float li = li_tm[ch];
    float st = state0[ch];
    for (int t = 0; t < T_DIM; ++t) {
      // s = s - s*li_tm + input (match reference op order)
      st = fmaf(-li, st, st) + cos_sh[t * 2 + ch];
      out[t * 2 + ch] = st;
    }
    out[T_DIM * 2 + ch] = st;  // final state
  }
}

extern "C" void kernel_launch(void* const* d_in, const int* in_sizes, int n_in,
                              void* d_out, int out_size, void* d_ws,
                              size_t ws_size, hipStream_t stream) {
  const float* x = (const float*)d_in[0];      // [512, 480, 640] f32
  const float* li_tm = (const float*)d_in[1];  // [2] f32
  const float* state = (const float*)d_in[2];  // [2] f32
  float* out = (float*)d_out;                  // [1026] f32
  float* ws = (float*)d_ws;                    // needs 512*10*2*4 = 40 KB

  dim3 grid(T_DIM, NSLICE);
  dsnt_partial_kernel<<<grid, 256, 0, stream>>>(x, ws);
  dsnt_scan_kernel<<<1, 256, 0, stream>>>(ws, li_tm, state, out);
}